// FusedAnsatz14_47493748359735
// MI455X (gfx1250) — compile-verified
//
#include <hip/hip_runtime.h>
#include <math.h>

// ---------------------------------------------------------------------------
// Fused 12-qubit ansatz simulator for MI455X (gfx1250, wave32).
//
// One workgroup per batch element. Full complex state (4096 amps, 32KB) lives
// in LDS for the entire 144-gate circuit. RY sublayers are fused per 4-qubit
// group into real 16x16 matrices and applied with V_WMMA_F32_16X16X4_F32
// chains (K=4 f32 WMMA, full f32 precision). CRX rings are VALU on LDS.
// ---------------------------------------------------------------------------

#define NQ      12
#define DIM     4096
#define LAYERS  3
#define NPARAM  144          // LAYERS*4*NQ
#define TPB     512          // 16 waves of 32

typedef __attribute__((ext_vector_type(2))) float v2f;
typedef __attribute__((ext_vector_type(8))) float v8f;

// State index for group-G 16x16 block application.
//  G=0: qubits 0-3  (bits 11..8): idx = k*256 + t*16 + n   (D = M*S, S 16x256)
//  G=1: qubits 4-7  (bits 7..4) : idx = t*256 + k*16 + n   (per-t 16x16 matmul)
//  G=2: qubits 8-11 (bits 3..0) : idx = t*256 + n*16 + k   (D = (M*S^T)^T)
template <int G>
__device__ __forceinline__ int sidx(int t, int k, int n) {
    if (G == 0) return k * 256 + t * 16 + n;
    if (G == 1) return t * 256 + k * 16 + n;
    return t * 256 + n * 16 + k;
}

// Apply group-G 16x16 matrix (mats[G*256..]) to both re/im state planes.
// 32 independent 16x16x16 matmuls (16 tiles x {re,im}); 16 waves -> 2 each.
// Each 16x16x16 product = 4 chained V_WMMA_F32_16X16X4_F32.
template <int G>
__device__ __forceinline__ void apply_group(float* sre, float* sim,
                                            const float* mats,
                                            int wave, int half, int l16) {
    const float* M = mats + G * 256;
#pragma unroll
    for (int q = 0; q < 2; ++q) {
        const int job = wave * 2 + q;       // 0..31
        const int t   = job >> 1;           // tile 0..15
        float* S = (job & 1) ? sim : sre;   // re / im plane

        v8f acc = {};
#pragma unroll
        for (int kb = 0; kb < 4; ++kb) {
            const int k0 = kb * 4 + half * 2;   // K columns for this lane half
            v2f a, bf;
            // A fragment: 16x4 f32 slice of M; lanes 0-15 rows, K {k0,k0+1},
            // lanes 16-31 same rows, K {k0,k0+1} with half*2 shift (ISA 7.12.2).
            a.x = M[l16 * 16 + k0];
            a.y = M[l16 * 16 + k0 + 1];
            // B fragment: 4x16 f32 slice of the state tile (row-striped).
            bf.x = S[sidx<G>(t, k0,     l16)];
            bf.y = S[sidx<G>(t, k0 + 1, l16)];
            acc = __builtin_amdgcn_wmma_f32_16x16x4_f32(
                false, a, false, bf, (short)0, acc, false, false);
        }
        // D write-back: VGPR r -> row r (lanes 0-15) / row r+8 (lanes 16-31).
#pragma unroll
        for (int r = 0; r < 8; ++r) {
            S[sidx<G>(t, r + half * 8, l16)] = acc[r];
        }
    }
}

// CRX(theta) with control bit pc, target bit pt (bit positions in the 4096
// index; qubit w lives at bit 11-w). Acts on control=1 subspace:
//   a' = c*a - i*s*b ; b' = -i*s*a + c*b   (a: tgt=0, b: tgt=1)
__device__ __forceinline__ void crx_gate(float* sre, float* sim,
                                         float c, float s,
                                         int pc, int pt, int tid) {
    const int lo = (pc < pt) ? pc : pt;
    const int hi = (pc < pt) ? pt : pc;
    for (int p = tid; p < 1024; p += TPB) {
        unsigned idx  = (unsigned)p;
        unsigned mlow = idx & ((1u << lo) - 1u);
        idx = ((idx >> lo) << (lo + 1)) | mlow;          // insert 0 at lo
        mlow = idx & ((1u << hi) - 1u);
        idx = ((idx >> hi) << (hi + 1)) | mlow;          // insert 0 at hi
        idx |= (1u << pc);                                // control = 1
        const unsigned i0 = idx;                          // target = 0
        const unsigned i1 = idx | (1u << pt);             // target = 1
        const float ar = sre[i0], ai = sim[i0];
        const float br = sre[i1], bi = sim[i1];
        sre[i0] = c * ar + s * bi;
        sim[i0] = c * ai - s * br;
        sre[i1] = c * br + s * ai;
        sim[i1] = c * bi - s * ar;
    }
}

__global__ __launch_bounds__(TPB, 1) void fused_ansatz_kernel(
    const float* __restrict__ g_re, const float* __restrict__ g_im,
    const float* __restrict__ g_par, float* __restrict__ out) {
    __shared__ float sre[DIM];        // 16 KB
    __shared__ float sim[DIM];        // 16 KB
    __shared__ float gc[NPARAM];      // cos(theta/2)
    __shared__ float gs[NPARAM];      // sin(theta/2)
    __shared__ float mats[3 * 256];   // three 16x16 RY-group matrices

    const int b    = blockIdx.x;
    const int tid  = threadIdx.x;
    const int lane = tid & 31;
    const int wave = tid >> 5;
    const int half = lane >> 4;
    const int l16  = lane & 15;

    // --- load params, precompute all cos/sin(theta/2) -----------------------
    for (int i = tid; i < NPARAM; i += TPB) {
        const float th = g_par[b * NPARAM + i] * 0.5f;
        gc[i] = cosf(th);
        gs[i] = sinf(th);
    }
    // --- load state into LDS ------------------------------------------------
    for (int i = tid; i < DIM; i += TPB) {
        sre[i] = g_re[b * DIM + i];
        sim[i] = g_im[b * DIM + i];
    }
    __syncthreads();

    for (int l = 0; l < LAYERS; ++l) {
#pragma unroll
        for (int sub = 0; sub < 2; ++sub) {   // sub 0: j=0 RY + ring1; sub 1: j=2 RY + ring2
            const int j = sub * 2;            // RY param row (0 or 2)

            // ---- RY sublayer: build 3 group matrices (tensor product of 4 RYs)
            for (int e = tid; e < 768; e += TPB) {
                const int g = e >> 8, rc = e & 255, r = rc >> 4, cc = rc & 15;
                float v = 1.0f;
#pragma unroll
                for (int t4 = 0; t4 < 4; ++t4) {
                    const int pi = (l * 4 + j) * 12 + (g * 4 + t4);
                    const int rb = (r >> (3 - t4)) & 1;
                    const int cb = (cc >> (3 - t4)) & 1;
                    // RY = [[c,-s],[s,c]]
                    v *= (rb == cb) ? gc[pi] : (rb ? gs[pi] : -gs[pi]);
                }
                mats[e] = v;
            }
            __syncthreads();

            apply_group<0>(sre, sim, mats, wave, half, l16);
            __syncthreads();
            apply_group<1>(sre, sim, mats, wave, half, l16);
            __syncthreads();
            apply_group<2>(sre, sim, mats, wave, half, l16);
            __syncthreads();

            // ---- CRX ring -------------------------------------------------
            if (sub == 0) {
                // gates: control w = 11..0, target (w+1)%12, angle index i=11-w
                for (int i = 0; i < NQ; ++i) {
                    const int w  = 11 - i;
                    const int pc = 11 - w;
                    const int pt = 11 - ((w + 1) % NQ);
                    const int pi = (l * 4 + 1) * 12 + i;
                    crx_gate(sre, sim, gc[pi], gs[pi], pc, pt, tid);
                    __syncthreads();
                }
            } else {
                // gates: w sequence [11,0,1,...,10], target (w-1)%12
                for (int i = 0; i < NQ; ++i) {
                    const int w  = (i == 0) ? 11 : (i - 1);
                    const int pc = 11 - w;
                    const int pt = 11 - ((w + 11) % NQ);
                    const int pi = (l * 4 + 3) * 12 + i;
                    crx_gate(sre, sim, gc[pi], gs[pi], pc, pt, tid);
                    __syncthreads();
                }
            }
        }
    }

    // --- write out [B, 4096, 2] (re, im stacked last) -----------------------
    for (int i = tid; i < DIM; i += TPB) {
        out[(b * DIM + i) * 2 + 0] = sre[i];
        out[(b * DIM + i) * 2 + 1] = sim[i];
    }
}

extern "C" void kernel_launch(void* const* d_in, const int* in_sizes, int n_in,
                              void* d_out, int out_size, void* d_ws, size_t ws_size,
                              hipStream_t stream) {
    const float* g_re  = (const float*)d_in[0];
    const float* g_im  = (const float*)d_in[1];
    const float* g_par = (const float*)d_in[2];
    float* out = (float*)d_out;
    const int B = in_sizes[0] / DIM;   // 128
    fused_ansatz_kernel<<<B, TPB, 0, stream>>>(g_re, g_im, g_par, out);
}